// RTD_30777735643331
// MI455X (gfx1250) — compile-verified
//
#include <hip/hip_runtime.h>

// ---------------------------------------------------------------------------
// RTD H0 total persistence, q = 1.
// Exact contraction: answer = MST total of {super-node S} U {Y_0..Y_4095}
//   S--Y_j        : c_j = min_{i<=j} XX[i][j]
//   Y_j--Y_k      : min(XX,YY)[j][k]
// Phase 1: fused f32-WMMA Gram GEMMs -> minXY matrix + c vector.
// Phase 2: persistent single-block Prim (4096 steps).
// ---------------------------------------------------------------------------

#define Nn 4096
#define Dd 256
#define BM 64
#define BN 32
#define KC 64
#define LDP 68   // padded LDS row stride (floats): bank-conflict-free, 16B aligned

typedef __attribute__((ext_vector_type(2))) float v2f;
typedef __attribute__((ext_vector_type(8))) float v8f;

#define FLTMAX_BITS 0x7F7FFFFFu

// ---------------------------------------------------------------------------
// Kernel 1: row norms for X and Y, init c (column-min accumulator) to FLT_MAX.
// One wave32 per row, coalesced lane-strided loads + cross-lane reduction.
// ---------------------------------------------------------------------------
__global__ __launch_bounds__(256) void rtd_norms_init(
    const float* __restrict__ X, const float* __restrict__ Y,
    float* __restrict__ nx, float* __restrict__ ny, unsigned* __restrict__ cbits) {
  int row  = (blockIdx.x * 256 + threadIdx.x) >> 5;
  int lane = threadIdx.x & 31;
  if (row >= Nn) return;
  const float* xr = X + (size_t)row * Dd;
  const float* yr = Y + (size_t)row * Dd;
  float sx = 0.f, sy = 0.f;
  for (int k = lane; k < Dd; k += 32) {
    float a = xr[k]; sx += a * a;
    float b = yr[k]; sy += b * b;
  }
  for (int off = 16; off; off >>= 1) {
    sx += __shfl_xor(sx, off, 32);
    sy += __shfl_xor(sy, off, 32);
  }
  if (lane == 0) {
    nx[row] = sx;
    ny[row] = sy;
    cbits[row] = FLTMAX_BITS;
  }
}

// ---------------------------------------------------------------------------
// Kernel 2: fused distance tiles via f32 WMMA.
//   Gx = X X^T tile, Gy = Y Y^T tile (16x16x4 f32 WMMA, K swept 256)
//   XX = sqrt(max(nx_i+nx_j-2Gx,0)); YY likewise
//   minXY[i][j] = min(XX,YY);  c_j = atomic-min over i<=j of XX[i][j]
// 256 threads = 8 waves, tile 64x32, each wave one 16x16 subtile.
// ---------------------------------------------------------------------------
__global__ __launch_bounds__(256) void rtd_dist_tiles(
    const float* __restrict__ X, const float* __restrict__ Y,
    const float* __restrict__ nx, const float* __restrict__ ny,
    float* __restrict__ minXY, unsigned* __restrict__ cbits) {
  __shared__ float Ax[BM * LDP];
  __shared__ float Bx[BN * LDP];
  __shared__ float Ay[BM * LDP];
  __shared__ float By[BN * LDP];
  __shared__ unsigned cmin[BN];

  const int tid  = threadIdx.x;
  const int lane = tid & 31;
  const int wv   = tid >> 5;
  const int subM = wv >> 1;   // 0..3
  const int subN = wv & 1;    // 0..1
  const int m0 = blockIdx.y * BM;
  const int n0 = blockIdx.x * BN;

  if (tid < BN) cmin[tid] = FLTMAX_BITS;

  v8f accX = {};
  v8f accY = {};

  const int hi  = lane >> 4;       // half-wave
  const int l15 = lane & 15;
  const int ra = (subM * 16 + l15) * LDP + (hi << 1);   // A fragment base (elems)
  const int rb = (subN * 16 + l15) * LDP + (hi << 1);   // B fragment base (elems)

  for (int k0 = 0; k0 < Dd; k0 += KC) {
    __syncthreads();
    // cooperative stage: A tiles (BM x KC) and B tiles (BN x KC), float4 loads
#pragma unroll
    for (int it = 0; it < 4; ++it) {
      int id = tid + 256 * it;
      int r = id >> 4, c4 = (id & 15) << 2;
      *(float4*)(&Ax[r * LDP + c4]) = *(const float4*)(X + (size_t)(m0 + r) * Dd + k0 + c4);
      *(float4*)(&Ay[r * LDP + c4]) = *(const float4*)(Y + (size_t)(m0 + r) * Dd + k0 + c4);
    }
#pragma unroll
    for (int it = 0; it < 2; ++it) {
      int id = tid + 256 * it;
      int r = id >> 4, c4 = (id & 15) << 2;
      *(float4*)(&Bx[r * LDP + c4]) = *(const float4*)(X + (size_t)(n0 + r) * Dd + k0 + c4);
      *(float4*)(&By[r * LDP + c4]) = *(const float4*)(Y + (size_t)(n0 + r) * Dd + k0 + c4);
    }
    __syncthreads();

#pragma unroll
    for (int kk = 0; kk < KC; kk += 4) {
      v2f a  = *(const v2f*)(&Ax[ra + kk]);
      v2f b  = *(const v2f*)(&Bx[rb + kk]);
      accX = __builtin_amdgcn_wmma_f32_16x16x4_f32(false, a, false, b,
                                                   (short)0, accX, false, false);
      v2f ay = *(const v2f*)(&Ay[ra + kk]);
      v2f by = *(const v2f*)(&By[rb + kk]);
      accY = __builtin_amdgcn_wmma_f32_16x16x4_f32(false, ay, false, by,
                                                   (short)0, accY, false, false);
    }
  }

  // epilogue: distances, minXY store, triangular column-min for c
  const int j     = n0 + subN * 16 + l15;          // global column
  const int iBase = m0 + subM * 16 + (hi << 3);    // global row base (VGPR e -> M=e[+8])
  const float nxj = nx[j];
  const float nyj = ny[j];
  float colmin = __uint_as_float(FLTMAX_BITS);
#pragma unroll
  for (int e = 0; e < 8; ++e) {
    int i = iBase + e;
    float xx = __builtin_sqrtf(fmaxf(nx[i] + nxj - 2.f * accX[e], 0.f));
    float yy = __builtin_sqrtf(fmaxf(ny[i] + nyj - 2.f * accY[e], 0.f));
    minXY[(size_t)i * Nn + j] = fminf(xx, yy);
    if (i <= j) colmin = fminf(colmin, xx);
  }
  // fold the two half-waves (same j, different i range), then LDS atomic-min
  colmin = fminf(colmin, __shfl_xor(colmin, 16, 32));
  if (lane < 16) atomicMin(&cmin[subN * 16 + lane], __float_as_uint(colmin));
  __syncthreads();
  if (tid < BN) atomicMin(&cbits[n0 + tid], cmin[tid]);  // uint order == float order (>=0)
}

// ---------------------------------------------------------------------------
// Kernel 3: Prim's MST on {S} U Y-nodes. Single block, 1024 threads.
// dist lives in registers (float4 per thread); argmin via packed u64 key
// (valbits<<32 | idx) -> lexicographic min == argmin with lowest index.
// ---------------------------------------------------------------------------
__global__ __launch_bounds__(1024) void rtd_prim(
    const float* __restrict__ minXY, const unsigned* __restrict__ cbits,
    float* __restrict__ out) {
  __shared__ unsigned long long red[32];
  __shared__ unsigned long long bkey;

  const int t = threadIdx.x;
  const float INF = __builtin_inff();

  uint4 cb = *(const uint4*)(cbits + 4 * t);
  float4 dist;
  dist.x = __uint_as_float(cb.x);
  dist.y = __uint_as_float(cb.y);
  dist.z = __uint_as_float(cb.z);
  dist.w = __uint_as_float(cb.w);
  unsigned visMask = 0;
  float total = 0.f;

  for (int step = 0; step < Nn; ++step) {
    float d0 = (visMask & 1u) ? INF : dist.x;
    float d1 = (visMask & 2u) ? INF : dist.y;
    float d2 = (visMask & 4u) ? INF : dist.z;
    float d3 = (visMask & 8u) ? INF : dist.w;
    float bv = d0; int bi = 4 * t;
    if (d1 < bv) { bv = d1; bi = 4 * t + 1; }
    if (d2 < bv) { bv = d2; bi = 4 * t + 2; }
    if (d3 < bv) { bv = d3; bi = 4 * t + 3; }
    unsigned long long key =
        ((unsigned long long)__float_as_uint(bv) << 32) | (unsigned)bi;

    // wave32 min-reduce
    for (int off = 16; off; off >>= 1) {
      unsigned long long o = __shfl_xor(key, off, 32);
      if (o < key) key = o;
    }
    if ((t & 31) == 0) red[t >> 5] = key;
    __syncthreads();
    if (t < 32) {
      unsigned long long k2 = red[t];
      for (int off = 16; off; off >>= 1) {
        unsigned long long o = __shfl_xor(k2, off, 32);
        if (o < k2) k2 = o;
      }
      if (t == 0) bkey = k2;
    }
    __syncthreads();

    unsigned long long k = bkey;
    int idx = (int)(unsigned)(k & 0xFFFFFFFFu);
    if (t == 0) total += fmaxf(__uint_as_float((unsigned)(k >> 32)), 0.f);
    if ((idx >> 2) == t) visMask |= 1u << (idx & 3);

    // relax with row idx of minXY (coalesced float4)
    float4 row = *(const float4*)(minXY + (size_t)idx * Nn + 4 * t);
    dist.x = fminf(dist.x, row.x);
    dist.y = fminf(dist.y, row.y);
    dist.z = fminf(dist.z, row.z);
    dist.w = fminf(dist.w, row.w);
  }
  if (t == 0) out[0] = total;
}

// ---------------------------------------------------------------------------
extern "C" void kernel_launch(void* const* d_in, const int* in_sizes, int n_in,
                              void* d_out, int out_size, void* d_ws, size_t ws_size,
                              hipStream_t stream) {
  (void)in_sizes; (void)n_in; (void)out_size; (void)ws_size;
  const float* X = (const float*)d_in[0];
  const float* Y = (const float*)d_in[1];
  float* out = (float*)d_out;

  // workspace layout: minXY (64 MB) | nx | ny | cbits
  float* minXY = (float*)d_ws;
  char* p = (char*)d_ws + (size_t)Nn * Nn * sizeof(float);
  float* nx = (float*)p;               p += Nn * sizeof(float);
  float* ny = (float*)p;               p += Nn * sizeof(float);
  unsigned* cbits = (unsigned*)p;

  rtd_norms_init<<<(Nn * 32) / 256, 256, 0, stream>>>(X, Y, nx, ny, cbits);

  dim3 grid(Nn / BN, Nn / BM);  // (128, 64)
  rtd_dist_tiles<<<grid, 256, 0, stream>>>(X, Y, nx, ny, minXY, cbits);

  rtd_prim<<<1, 1024, 0, stream>>>(minXY, cbits, out);
}